// OrthogonalConstraint_42314017800809
// MI455X (gfx1250) — compile-verified
//
#include <hip/hip_runtime.h>

// Batched modified Gram-Schmidt, (4096, 8, 768) f32, reformulated as:
//   G = V V^T  (WMMA f32 16x16x4, K=768)          -> 8x8 Gram per batch
//   coefficient-space MGS recurrence on G (scalar, 2 lanes, LDS, kept rolled)
//   Q = C V    (WMMA f32 16x16x4, N tiled by 16)
// 2 batch elements per wave (16 WMMA rows, block-diagonal C), 1 wave per WG.

typedef float v2f __attribute__((ext_vector_type(2)));
typedef float v8f __attribute__((ext_vector_type(8)));

#define GS_EPS 1e-6f
#define GS_S 8
#define GS_D 768

__global__ __launch_bounds__(32)
void OrthogonalConstraint_gs_kernel(const float* __restrict__ in,
                                    float* __restrict__ out)
{
    __shared__ float sG[256];   // 16x16 Gram (two 8x8 diagonal blocks valid)
    __shared__ float sC[256];   // 16x16 block-diagonal coefficient matrix
    __shared__ float sQQ[16];   // per-row q.q denominators
    __shared__ float sR[16];    // residual coefficient scratch (2 x 8)

    const int lane  = threadIdx.x & 31;
    const int row16 = lane & 15;     // WMMA M (or N) index
    const int hi    = lane >> 4;     // K half select

    const size_t pairBase = (size_t)blockIdx.x * (size_t)(2 * GS_S * GS_D);
    const float* V = in  + pairBase; // 16 rows x 768 (batch0 rows 0-7, batch1 rows 8-15)
    float*       Q = out + pairBase;

    // Warm the cache hierarchy for the whole 48KB pair-tile: the phase-1
    // fragment loads are 16-row strided, so pull full lines in first.
    // 32 lanes x 128B x 12 steps = 48KB  (global_prefetch_b8).
    #pragma unroll
    for (int it = 0; it < 12; ++it)
        __builtin_prefetch(V + ((size_t)(it * 32 + lane) * 32), 0, 1);

    // ---------------- Phase 1: G = V * V^T -------------------------------
    // A-frag lane layout: A[M=row16][K = 4*kc + 2*hi + {0,1}]
    // B-frag lane layout: B[K = 4*kc + 2*hi + {0,1}][N=row16] = V[row16][K]
    // => identical register image for the symmetric product.
    v8f acc = {};
    const float* arow = V + (size_t)row16 * GS_D;
    #pragma unroll 4
    for (int kc = 0; kc < GS_D / 4; ++kc) {
        const int k = kc * 4 + 2 * hi;
        v2f a = *(const v2f*)(arow + k);       // b64, 8B aligned
        acc = __builtin_amdgcn_wmma_f32_16x16x4_f32(false, a, false, a,
                                                    (short)0, acc, false, false);
    }

    // D layout: lane L, vgpr r -> D[M = r + 8*hi][N = row16]
    #pragma unroll
    for (int r = 0; r < 8; ++r)
        sG[(r + 8 * hi) * 16 + row16] = acc[r];

    // zero the block-diagonal coefficient scratch (256 entries / 32 lanes)
    #pragma unroll
    for (int r = 0; r < 8; ++r)
        sC[r * 32 + lane] = 0.0f;

    __syncthreads();

    // ---------------- Phase 2: coefficient-space MGS ----------------------
    // lane 0 -> batch 0 block, lane 1 -> batch 1 block. Mirrors the reference
    // loop exactly (MGS residual update order, denom = q_j.q_j, subtract
    // gated on sqrt(denom) > EPS, divide gated on norm > EPS). Executed once
    // by 2 lanes -> keep it ROLLED (tiny I-footprint), state lives in LDS.
    if (lane < 2) {
        const int o = 8 * lane;                 // block offset in 16x16
        #pragma clang loop unroll(disable)
        for (int i = 0; i < 8; ++i) {
            #pragma clang loop unroll(disable)
            for (int k = 0; k < 8; ++k) sR[o + k] = (k == i) ? 1.0f : 0.0f;
            #pragma clang loop unroll(disable)
            for (int j = 0; j < i; ++j) {
                const float denom = sQQ[o + j];
                const float safe  = (denom > 0.0f) ? denom : 1.0f;
                float dot = 0.0f;
                #pragma clang loop unroll(disable)
                for (int k = 0; k < 8; ++k) {
                    float gc = 0.0f;
                    #pragma clang loop unroll(disable)
                    for (int l = 0; l < 8; ++l)
                        gc += sG[(k + o) * 16 + (l + o)] * sC[(j + o) * 16 + (l + o)];
                    dot += sR[o + k] * gc;
                }
                const float proj = dot / safe;
                if (__builtin_sqrtf(denom) > GS_EPS) {
                    #pragma clang loop unroll(disable)
                    for (int k = 0; k < 8; ++k)
                        sR[o + k] -= proj * sC[(j + o) * 16 + (k + o)];
                }
            }
            float n2 = 0.0f;
            #pragma clang loop unroll(disable)
            for (int k = 0; k < 8; ++k) {
                float gr = 0.0f;
                #pragma clang loop unroll(disable)
                for (int l = 0; l < 8; ++l)
                    gr += sG[(k + o) * 16 + (l + o)] * sR[o + l];
                n2 += sR[o + k] * gr;
            }
            n2 = (n2 > 0.0f) ? n2 : 0.0f;
            const float norm = __builtin_sqrtf(n2);
            const float inv  = (norm > GS_EPS) ? (1.0f / norm) : 1.0f;
            #pragma clang loop unroll(disable)
            for (int k = 0; k < 8; ++k)
                sC[(i + o) * 16 + (k + o)] = sR[o + k] * inv;
            sQQ[o + i] = n2 * inv * inv;        // dot(q_i, q_i)
        }
    }
    __syncthreads();

    // ---------------- Phase 3: Q = C * V (N tiles of 16) ------------------
    // A-frag (C) is constant across N tiles: load once from LDS.
    v2f afrag[4];
    #pragma unroll
    for (int kc = 0; kc < 4; ++kc) {
        const int k = kc * 4 + 2 * hi;
        afrag[kc].x = sC[row16 * 16 + k];
        afrag[kc].y = sC[row16 * 16 + k + 1];
    }

    for (int n0 = 0; n0 < GS_D; n0 += 16) {
        v8f q = {};
        #pragma unroll
        for (int kc = 0; kc < 4; ++kc) {
            const int kf = kc * 4 + 2 * hi;
            v2f b;                               // B[K][N=row16] = V[K][n0+row16]
            b.x = V[(size_t)kf * GS_D + n0 + row16];
            b.y = V[(size_t)(kf + 1) * GS_D + n0 + row16];
            q = __builtin_amdgcn_wmma_f32_16x16x4_f32(false, afrag[kc], false, b,
                                                      (short)0, q, false, false);
        }
        #pragma unroll
        for (int r = 0; r < 8; ++r)
            Q[(size_t)(r + 8 * hi) * GS_D + n0 + row16] = q[r];
    }
}

extern "C" void kernel_launch(void* const* d_in, const int* in_sizes, int n_in,
                              void* d_out, int out_size, void* d_ws, size_t ws_size,
                              hipStream_t stream) {
    (void)n_in; (void)out_size; (void)d_ws; (void)ws_size;
    const float* in  = (const float*)d_in[0];
    float*       out = (float*)d_out;
    const int nBatch = in_sizes[0] / (GS_S * GS_D);   // 4096
    const int nBlocks = nBatch / 2;                   // 2 batches per wave/WG
    OrthogonalConstraint_gs_kernel<<<nBlocks, 32, 0, stream>>>(in, out);
}